// RGCNHighMem_4475355922762
// MI455X (gfx1250) — compile-verified
//
#include <hip/hip_runtime.h>

typedef float v2f __attribute__((ext_vector_type(2)));
typedef float v8f __attribute__((ext_vector_type(8)));

#define IN_FEAT  32
#define OUT_FEAT 32
#define NUM_RELS 64
#define TILE     16

// ---------------- init: zero output, init workspace ----------------
__global__ void init_kernel(float* __restrict__ out, int out_n,
                            int* __restrict__ counts, int* __restrict__ cursors,
                            int* __restrict__ tile_rel, int max_tiles,
                            int* __restrict__ sorted, int e_pad, int total) {
    int i = blockIdx.x * blockDim.x + threadIdx.x;
    int stride = gridDim.x * blockDim.x;
    for (int idx = i; idx < total; idx += stride) {
        if (idx < out_n)     out[idx] = 0.0f;
        if (idx < NUM_RELS)  { counts[idx] = 0; cursors[idx] = 0; }
        if (idx < max_tiles) tile_rel[idx] = -1;
        if (idx < e_pad)     sorted[idx] = -1;
    }
}

// ---------------- histogram over relations ----------------
__global__ void hist_kernel(const int* __restrict__ etypes, int n_edges,
                            int* __restrict__ counts) {
    int i = blockIdx.x * blockDim.x + threadIdx.x;
    if (i < n_edges) atomicAdd(&counts[etypes[i]], 1);
}

// ---------------- padded exclusive scan + tile->relation map ----------------
__global__ void scan_kernel(const int* __restrict__ counts,
                            int* __restrict__ poff,
                            int* __restrict__ tile_rel) {
    __shared__ int scnt[NUM_RELS];
    __shared__ int soff[NUM_RELS];
    int r = threadIdx.x;
    scnt[r] = counts[r];
    __syncthreads();
    if (r == 0) {
        int acc = 0;
        for (int i = 0; i < NUM_RELS; ++i) {
            soff[i] = acc;
            acc += ((scnt[i] + TILE - 1) / TILE) * TILE;  // pad to tile multiple
        }
    }
    __syncthreads();
    poff[r] = soff[r];
    int base_tile = soff[r] / TILE;
    int ntile = (scnt[r] + TILE - 1) / TILE;
    for (int i = 0; i < ntile; ++i) tile_rel[base_tile + i] = r;
}

// ---------------- scatter edge ids into padded sorted order ----------------
__global__ void scatter_kernel(const int* __restrict__ etypes, int n_edges,
                               const int* __restrict__ poff,
                               int* __restrict__ cursors,
                               int* __restrict__ sorted) {
    int i = blockIdx.x * blockDim.x + threadIdx.x;
    if (i < n_edges) {
        int r = etypes[i];
        int pos = poff[r] + atomicAdd(&cursors[r], 1);
        sorted[pos] = i;
    }
}

// ---------------- WMMA tile kernel: one wave per 16-edge tile ----------------
// A (16x4 f32): lane L holds rows M=L&15; VGPR0 = K=c0, VGPR1 = K=c0+1, c0 = (L<16?0:2)
// B (4x16 f32): lane L holds cols N=L&15; VGPR0 = K=c0 row, VGPR1 = K=c0+1 row
// C/D (16x16):  VGPR j: lanes 0-15 -> M=j, lanes 16-31 -> M=j+8; N = L&15
__global__ __launch_bounds__(256) void rgcn_wmma_kernel(
    const float* __restrict__ feat, const float* __restrict__ weight,
    const int* __restrict__ src, const int* __restrict__ dst,
    const int* __restrict__ sorted, const int* __restrict__ tile_rel,
    float* __restrict__ out, int max_tiles) {

    int wave = threadIdx.x >> 5;
    int t = blockIdx.x * 8 + wave;
    if (t >= max_tiles) return;          // wave-uniform
    int r = tile_rel[t];
    if (r < 0) return;                   // wave-uniform (empty padded tile)

    int lane  = threadIdx.x & 31;
    int m     = lane & 15;
    int khalf = lane >> 4;               // 0 -> K base 0, 1 -> K base 2
    int c0    = khalf * 2;

    // This lane's A-row edge (row m of the tile)
    int e = sorted[t * TILE + m];
    float valf = (e >= 0) ? 1.0f : 0.0f; // branch-free masking keeps EXEC full
    int sn = (e >= 0) ? src[e] : 0;
    const float* fsrc = feat + (size_t)sn * IN_FEAT;
    const float* W    = weight + (size_t)r * IN_FEAT * OUT_FEAT;

    v8f c_lo = {};   // output columns 0..15
    v8f c_hi = {};   // output columns 16..31

#pragma unroll
    for (int s = 0; s < 8; ++s) {        // K = 32 in steps of 4
        int kb = s * 4 + c0;
        v2f a = *(const v2f*)(fsrc + kb);  // 8B-aligned (kb even)
        a.x *= valf; a.y *= valf;

        v2f b0, b1;
        b0.x = W[(size_t)kb * OUT_FEAT + m];
        b0.y = W[(size_t)(kb + 1) * OUT_FEAT + m];
        b1.x = W[(size_t)kb * OUT_FEAT + 16 + m];
        b1.y = W[(size_t)(kb + 1) * OUT_FEAT + 16 + m];

        c_lo = __builtin_amdgcn_wmma_f32_16x16x4_f32(
            false, a, false, b0, (short)0, c_lo, false, false);
        c_hi = __builtin_amdgcn_wmma_f32_16x16x4_f32(
            false, a, false, b1, (short)0, c_hi, false, false);
    }

    // Scatter-add rows to destination nodes (native f32 atomics).
#pragma unroll
    for (int j = 0; j < 8; ++j) {
        int row = j + khalf * 8;
        int e2 = sorted[t * TILE + row];
        if (e2 >= 0) {
            int d = dst[e2];
            float* op = out + (size_t)d * OUT_FEAT;
            unsafeAtomicAdd(op + m,      c_lo[j]);
            unsafeAtomicAdd(op + 16 + m, c_hi[j]);
        }
    }
}

// ---------------- launch ----------------
extern "C" void kernel_launch(void* const* d_in, const int* in_sizes, int n_in,
                              void* d_out, int out_size, void* d_ws, size_t ws_size,
                              hipStream_t stream) {
    const float* feat   = (const float*)d_in[0];
    const float* weight = (const float*)d_in[1];
    const int*   src    = (const int*)d_in[2];
    const int*   dst    = (const int*)d_in[3];
    const int*   etypes = (const int*)d_in[4];
    float* out = (float*)d_out;

    int n_edges  = in_sizes[2];
    int max_tiles = (n_edges + TILE - 1) / TILE + NUM_RELS;  // worst-case padding
    int e_pad     = max_tiles * TILE;

    int* counts   = (int*)d_ws;              // [64]
    int* poff     = counts + NUM_RELS;       // [64] padded segment starts (edges)
    int* cursors  = poff + NUM_RELS;         // [64]
    int* tile_rel = cursors + NUM_RELS;      // [max_tiles]
    int* sorted   = tile_rel + max_tiles;    // [e_pad]

    int init_total = out_size > e_pad ? out_size : e_pad;
    init_kernel<<<(init_total + 255) / 256, 256, 0, stream>>>(
        out, out_size, counts, cursors, tile_rel, max_tiles, sorted, e_pad, init_total);
    hist_kernel<<<(n_edges + 255) / 256, 256, 0, stream>>>(etypes, n_edges, counts);
    scan_kernel<<<1, NUM_RELS, 0, stream>>>(counts, poff, tile_rel);
    scatter_kernel<<<(n_edges + 255) / 256, 256, 0, stream>>>(
        etypes, n_edges, poff, cursors, sorted);
    rgcn_wmma_kernel<<<(max_tiles + 7) / 8, 256, 0, stream>>>(
        feat, weight, src, dst, sorted, tile_rel, out, max_tiles);
}